// HeterogeneousGNN_91104846283471
// MI455X (gfx1250) — compile-verified
//
#include <hip/hip_runtime.h>

// ---------------------------------------------------------------------------
// HeterogeneousGNN on MI455X (gfx1250, wave32, WMMA).
//
// Roofline: mandatory HBM traffic ~370MB -> ~16us @ 23.3 TB/s. Compute is
// ~21 GFLOP -> negligible even with pure-f32 WMMA (v_wmma_f32_16x16x4_f32),
// so we keep full f32 precision (matches reference numerics).
//
// L2 strategy (192 MB): obj_lin (25.6MB) + out accumulator (25.6MB) + weights
// must stay L2-resident so edge gathers and the 76.8M f32 atomics resolve at
// L2. rel_vecs (307MB), obj_vecs and edge_index are one-shot streams -> load
// them NON-TEMPORAL (TH_LOAD_NT) so they don't evict the hot working set.
// The fused edge kernel never materializes rel_lin/msg (saves ~600MB HBM).
// ---------------------------------------------------------------------------

#define N_NODES 50000
#define N_EDGES 600000
#define DIM 128
#define LDS_STRIDE 132   // pad 128->132 floats: lanes 0..15 hit distinct banks

typedef float v2f __attribute__((ext_vector_type(2)));
typedef float v4f __attribute__((ext_vector_type(4)));
typedef float v8f __attribute__((ext_vector_type(8)));

// D(16x16) += A(16x4,f32) x B(4x16,f32), K accumulated in steps of 4.
// A-frag (ISA 7.12.2, 32-bit A 16x4): lane l<16 -> (M=l, K=4s+0..1),
//                                     lane l>=16 -> (M=l-16, K=4s+2..3).
// C/D: VGPR r -> M=r (lanes 0-15) / M=r+8 (lanes 16-31), N = lane%16.
__device__ __forceinline__ v8f wmma_k128(const float* __restrict__ Arow, // LDS, one M-row base
                                         const float* __restrict__ Brow, // global, L2-hot weights
                                         int hi, v8f c)
{
#pragma unroll
    for (int s = 0; s < 32; ++s) {
        const int kb = 4 * s + 2 * hi;
        v2f a = *(const v2f*)(Arow + kb);   // ds_load_b64
        v2f b = *(const v2f*)(Brow + kb);   // global b64, RT hint (keep in L2)
        c = __builtin_amdgcn_wmma_f32_16x16x4_f32(false, a, false, b,
                                                  (short)0, c, false, false);
    }
    return c;
}

// Stage a 16x128 f32 tile into LDS with non-temporal loads (one-shot stream).
__device__ __forceinline__ void stage_tile_nt(const float* __restrict__ gsrc,
                                              float* __restrict__ As, int t)
{
    const v4f* src = (const v4f*)gsrc;
#pragma unroll
    for (int i = 0; i < 2; ++i) {
        int idx  = t + i * 256;
        int row  = idx >> 5;                 // 32 float4 per row
        int col4 = idx & 31;
        v4f v = __builtin_nontemporal_load(&src[idx]);   // b128, TH_LOAD_NT
        float* d = &As[row * LDS_STRIDE + col4 * 4];
        d[0] = v.x; d[1] = v.y; d[2] = v.z; d[3] = v.w;
    }
}

// Y[r, n] = sum_k X[r,k] * W[n,k] + b[n]      (rows % 16 == 0)
__global__ void __launch_bounds__(256) gemm_bias_kernel(const float* __restrict__ X,
                                                        const float* __restrict__ W,
                                                        const float* __restrict__ bias,
                                                        float* __restrict__ Y)
{
    __shared__ float As[16 * LDS_STRIDE];
    const int t = threadIdx.x;
    const long long rowBase = (long long)blockIdx.x * 16;

    stage_tile_nt(X + rowBase * DIM, As, t);
    __syncthreads();

    const int wave = t >> 5;
    const int lane = t & 31;
    const int l    = lane & 15;
    const int hi   = lane >> 4;
    const int n0   = wave * 16;        // each of 8 waves owns 16 output cols

    v8f c = {};
    c = wmma_k128(&As[l * LDS_STRIDE], W + (long long)(n0 + l) * DIM, hi, c);

    const int col = n0 + l;
    const float bv = bias[col];
#pragma unroll
    for (int r = 0; r < 8; ++r) {
        long long row = rowBase + r + 8 * hi;
        Y[row * DIM + col] = c[r] + bv;    // RT store: wants to live in L2
    }
}

// Fused: rel_lin GEMM + gather(obj_lin[src], obj_lin[dst]) + relu + atomic scatter.
__global__ void __launch_bounds__(256) edge_fused_kernel(const float* __restrict__ relv,
                                                         const int* __restrict__ eidx,
                                                         const float* __restrict__ Wrel,
                                                         const float* __restrict__ brel,
                                                         const float* __restrict__ objlin,
                                                         float* __restrict__ out)
{
    __shared__ float As[16 * LDS_STRIDE];
    __shared__ int sSrc[16];
    __shared__ int sDst[16];
    const int t = threadIdx.x;
    const long long eBase = (long long)blockIdx.x * 16;

    if (t < 16)       sSrc[t]      = __builtin_nontemporal_load(&eidx[2 * (eBase + t)]);
    else if (t < 32)  sDst[t - 16] = __builtin_nontemporal_load(&eidx[2 * (eBase + (t - 16)) + 1]);

    stage_tile_nt(relv + eBase * DIM, As, t);   // 307MB stream: NT, don't pollute L2
    __syncthreads();

    const int wave = t >> 5;
    const int lane = t & 31;
    const int l    = lane & 15;
    const int hi   = lane >> 4;
    const int n0   = wave * 16;

    v8f c = {};
    c = wmma_k128(&As[l * LDS_STRIDE], Wrel + (long long)(n0 + l) * DIM, hi, c);

    const int col = n0 + l;
    const float bv = brel[col];
#pragma unroll
    for (int r = 0; r < 8; ++r) {
        int er = r + 8 * hi;                 // edge within tile (C/D row map)
        int sN = sSrc[er];
        int dN = sDst[er];
        float v = c[r] + bv
                + objlin[(long long)sN * DIM + col]    // L2 hit (25.6MB table)
                + objlin[(long long)dN * DIM + col];   // L2 hit
        v = v > 0.0f ? v : 0.0f;
        atomicAdd(&out[(long long)dN * DIM + col], v); // global_atomic_add_f32 @ L2
    }
}

__global__ void __launch_bounds__(256) zero_kernel(float4* __restrict__ p, int n4)
{
    int i = blockIdx.x * blockDim.x + threadIdx.x;
    if (i < n4) p[i] = make_float4(0.f, 0.f, 0.f, 0.f);
}

__global__ void __launch_bounds__(256) relu_kernel(float4* __restrict__ p, int n4)
{
    int i = blockIdx.x * blockDim.x + threadIdx.x;
    if (i < n4) {
        float4 v = p[i];
        v.x = fmaxf(v.x, 0.f); v.y = fmaxf(v.y, 0.f);
        v.z = fmaxf(v.z, 0.f); v.w = fmaxf(v.w, 0.f);
        p[i] = v;
    }
}

extern "C" void kernel_launch(void* const* d_in, const int* in_sizes, int n_in,
                              void* d_out, int out_size, void* d_ws, size_t ws_size,
                              hipStream_t stream)
{
    const float* obj_vecs = (const float*)d_in[0];
    const float* rel_vecs = (const float*)d_in[1];
    const int*   edge_idx = (const int*)  d_in[2];
    const float* W_obj    = (const float*)d_in[3];
    const float* b_obj    = (const float*)d_in[4];
    const float* W_rel    = (const float*)d_in[5];
    const float* b_rel    = (const float*)d_in[6];

    float* out     = (float*)d_out;
    float* obj_lin = (float*)d_ws;            // 50000*128*4 = 25.6 MB scratch

    // 1) obj_lin = obj_vecs @ W_obj.T + b_obj   (L2-resident afterwards)
    gemm_bias_kernel<<<N_NODES / 16, 256, 0, stream>>>(obj_vecs, W_obj, b_obj, obj_lin);

    // 2) zero the accumulator (harness poisons d_out)
    const int n4 = N_NODES * DIM / 4;
    zero_kernel<<<(n4 + 255) / 256, 256, 0, stream>>>((float4*)out, n4);

    // 3) fused rel GEMM + gather + relu + scatter-add
    edge_fused_kernel<<<N_EDGES / 16, 256, 0, stream>>>(rel_vecs, edge_idx, W_rel,
                                                        b_rel, obj_lin, out);

    // 4) final relu in place
    relu_kernel<<<(n4 + 255) / 256, 256, 0, stream>>>((float4*)out, n4);
}